// Seq_Struce_Bert_11123965297247
// MI455X (gfx1250) — compile-verified
//
#include <hip/hip_runtime.h>
#include <hip/hip_bf16.h>
#include <math.h>

// ---------------------------------------------------------------------------
// Seq_Struct_BERT forward for MI455X (gfx1250, wave32, WMMA + TDM).
// All GEMMs use V_WMMA_F32_16X16X4_F32 (true fp32 MACs -> matches fp32 ref).
// Weight panels are staged to LDS with TENSOR_LOAD_TO_LDS (async tensor DMA).
// ---------------------------------------------------------------------------

typedef __attribute__((ext_vector_type(2))) float v2f;
typedef __attribute__((ext_vector_type(8))) float v8f;
typedef __attribute__((ext_vector_type(4))) unsigned int u32x4;
typedef __attribute__((ext_vector_type(8))) int i32x8;
typedef __attribute__((ext_vector_type(4))) int i32x4;

#define BDIM 4
#define SDIM 2048
#define EDIM 512
#define FDIM 16
#define DFF  2048
#define NHEAD 8
#define DH    64
#define MROWS (BDIM * SDIM)   // 8192
#define NMASK 256
#define KC    32              // GEMM K-chunk staged in LDS

#if defined(__has_builtin)
#if __has_builtin(__builtin_amdgcn_tensor_load_to_lds) && \
    __has_builtin(__builtin_amdgcn_s_wait_tensorcnt)
#define USE_TDM 1
#endif
#endif

__device__ __forceinline__ v8f wmma4_f32(v2f a, v2f b, v8f c) {
    // D = A(16x4) * B(4x16) + C, fp32 throughout.
    return __builtin_amdgcn_wmma_f32_16x16x4_f32(
        false, a, false, b, (short)0, c, false, false);
}

__device__ __forceinline__ v8f zero8() {
    v8f z = {0.f, 0.f, 0.f, 0.f, 0.f, 0.f, 0.f, 0.f};
    return z;
}

#ifdef USE_TDM
// Byte offset of a generic pointer within LDS (AS3 address == LDS offset).
__device__ __forceinline__ unsigned int lds_addr_of(const void* p) {
    return (unsigned int)(unsigned long long)
        (__attribute__((address_space(3))) const void*)p;
}

// One 2D TDM transfer: tile of KC rows x 64 f32 cols from row-major
// global (row stride = stride_elems) into contiguous LDS at ldsOff.
// D# per CDNA5 ISA. EXEC-independent; issue
// from one wave per workgroup, then s_wait_tensorcnt(0).
// This toolchain exposes the 6-arg builtin:
//   (u32x4 g0, i32x8 g1, i32x4 g2, i32x4 g3, i32x8 g4, i32 cpol)
__device__ __forceinline__ void tdm_load_tile(const void* gptr,
                                              unsigned int ldsOff,
                                              int stride_elems) {
    unsigned long long ga = (unsigned long long)gptr;
    // group0: count=1 | lds_addr | global_addr[56:0] | type=2
    u32x4 g0 = { 1u,
                 ldsOff,
                 (unsigned int)ga,
                 (unsigned int)(ga >> 32) | (2u << 30) };
    // group1: data_size=4B; tensor_dim0=64, tensor_dim1=KC;
    //         tile_dim0=64, tile_dim1=KC; tensor_dim0_stride=stride_elems.
    i32x8 g1 = { (int)(2u << 16),            // data_size=2 (4 bytes)
                 (int)(64u << 16),           // tensor_dim0[15:0]
                 (int)((unsigned)KC << 16),  // tensor_dim1[15:0]
                 (int)(64u << 16),           // tile_dim0
                 (int)KC,                    // tile_dim1 (tile_dim2=0)
                 stride_elems,               // tensor_dim0_stride[31:0]
                 0, 0 };
    i32x4 gz4 = { 0, 0, 0, 0 };
    i32x8 gz8 = { 0, 0, 0, 0, 0, 0, 0, 0 };
    __builtin_amdgcn_tensor_load_to_lds(g0, g1, gz4, gz4, gz8, 0);
}
#endif

// ---------------------------------------------------------------------------
// Preprocess kernels
// ---------------------------------------------------------------------------

// mclass: 0 = padded, 1 = valid, 2 = masked position. Also emit out mask (f32).
__global__ void mask_init_kernel(const unsigned char* __restrict__ mask,
                                 float* __restrict__ mclass,
                                 float* __restrict__ out_mask) {
    int i = blockIdx.x * blockDim.x + threadIdx.x;
    if (i < BDIM * SDIM) {
        unsigned char m = mask[i];
        mclass[i]   = m ? 0.f : 1.f;
        out_mask[i] = m ? 1.f : 0.f;
    }
}

__global__ void scatter_masked_kernel(const float* __restrict__ x,
                                      const int* __restrict__ masked_pos,
                                      float* __restrict__ mclass,
                                      float* __restrict__ label) {
    int idx = blockIdx.x * blockDim.x + threadIdx.x;   // 0 .. 1023
    if (idx < BDIM * NMASK) {
        int b = idx >> 8;
        int p = masked_pos[idx];
        mclass[b * SDIM + p] = 2.f;
        const float* xr = x + ((size_t)b * SDIM + p) * FDIM;
        float* lr = label + (size_t)idx * FDIM;
#pragma unroll
        for (int f = 0; f < FDIM; ++f) lr[f] = xr[f];
    }
}

// One 128-thread block per token: embedding lookup + min/max normalize,
// concat with first 15 features (zeroed if masked), add positional encoding.
__global__ __launch_bounds__(128)
void build_h0_kernel(const float* __restrict__ x,
                     const float* __restrict__ mclass,
                     const float* __restrict__ emb_table,   // (20, 497)
                     const float* __restrict__ pe,          // (2048, 512)
                     float* __restrict__ h) {
    __shared__ float smin[4], smax[4];
    int tok = blockIdx.x;                 // 0 .. 8191
    int s = tok & (SDIM - 1);
    int t = threadIdx.x;
    int lane = t & 31, wv = t >> 5;

    bool masked = (mclass[tok] == 2.f);
    const float* xr = x + (size_t)tok * FDIM;
    int amino = masked ? 0 : (int)xr[FDIM - 1];
    const float* et = emb_table + (size_t)amino * (EDIM - 15);

    float mn = INFINITY, mx = -INFINITY;
    for (int i = t; i < EDIM - 15; i += 128) {
        float e = et[i];
        mn = fminf(mn, e);
        mx = fmaxf(mx, e);
    }
#pragma unroll
    for (int off = 16; off > 0; off >>= 1) {
        mn = fminf(mn, __shfl_xor(mn, off, 32));
        mx = fmaxf(mx, __shfl_xor(mx, off, 32));
    }
    if (lane == 0) { smin[wv] = mn; smax[wv] = mx; }
    __syncthreads();
    mn = fminf(fminf(smin[0], smin[1]), fminf(smin[2], smin[3]));
    mx = fmaxf(fmaxf(smax[0], smax[1]), fmaxf(smax[2], smax[3]));
    float inv = 1.f / (mx - mn);

    float* hr = h + (size_t)tok * EDIM;
    const float* per = pe + (size_t)s * EDIM;
    for (int i = t; i < EDIM; i += 128) {
        float val;
        if (i < 15) val = masked ? 0.f : xr[i];
        else        val = (et[i - 15] - mn) * inv;
        hr[i] = val + per[i];
    }
}

// ---------------------------------------------------------------------------
// fp32 WMMA GEMM: D(M,N) = A(M,K) @ W(K,N) + bias [+ resid] [relu]
// block = 256 threads = 8 waves; wave -> 16x64 strip; WG tile = 128 x 64.
// W panels (KC x 64) staged to LDS via TDM (wave 0 issues; TENSORcnt wait),
// shared by all 8 waves. A rows streamed from global as float2 per lane.
// Grids are exact (no early return: barriers inside the K loop).
// ---------------------------------------------------------------------------
template <bool RELU, bool RESID>
__global__ __launch_bounds__(256)
void gemm_f32_kernel(const float* __restrict__ A, const float* __restrict__ W,
                     const float* __restrict__ bias,
                     const float* __restrict__ resid,
                     float* __restrict__ D, int M, int N, int K) {
    __shared__ float ldsB[KC * 64];
    const int wave = threadIdx.x >> 5;
    const int lane = threadIdx.x & 31;
    const int half = lane >> 4;
    const int l16  = lane & 15;
    const int m0 = (blockIdx.y * 8 + wave) * 16;
    const int n0 = blockIdx.x * 64;

    v8f acc[4];
#pragma unroll
    for (int t = 0; t < 4; ++t) acc[t] = zero8();

    const float* arow = A + (size_t)(m0 + l16) * K + half * 2;

    for (int kc = 0; kc < K; kc += KC) {
#ifdef USE_TDM
        if (wave == 0) {
            tdm_load_tile(W + (size_t)kc * N + n0, lds_addr_of(ldsB), N);
            __builtin_amdgcn_s_wait_tensorcnt(0);
        }
#else
        for (int i = threadIdx.x; i < KC * 64; i += 256)
            ldsB[i] = W[(size_t)(kc + (i >> 6)) * N + n0 + (i & 63)];
#endif
        __syncthreads();
        if (kc + KC < K)
            __builtin_prefetch(arow + kc + KC, 0, 0);   // global_prefetch_b8

#pragma unroll
        for (int kk = 0; kk < KC; kk += 4) {
            v2f a = *(const v2f*)(arow + kc + kk);
            const float* bk = ldsB + (kk + half * 2) * 64;
#pragma unroll
            for (int t = 0; t < 4; ++t) {
                int c = t * 16 + l16;
                v2f b;
                b.x = bk[c];
                b.y = bk[c + 64];
                acc[t] = wmma4_f32(a, b, acc[t]);
            }
        }
        __syncthreads();   // protect ldsB before next chunk's DMA
    }

#pragma unroll
    for (int t = 0; t < 4; ++t) {
        int col = n0 + t * 16 + l16;
        float bv = bias[col];
#pragma unroll
        for (int j = 0; j < 8; ++j) {
            int row = m0 + j + half * 8;
            float v = acc[t][j] + bv;
            if (RESID) v += resid[(size_t)row * N + col];
            if (RELU)  v = v > 0.f ? v : 0.f;
            D[(size_t)row * N + col] = v;
        }
    }
}

// ---------------------------------------------------------------------------
// Attention: flash-style online softmax. One wave per 16-query tile per (b,h).
// q/k/v are (B,S,E) with head h occupying columns [h*64, h*64+64).
// scores = (q.k^T)/Dh * m  (m==0 -> -inf), softmax over keys, out = P @ v.
// ---------------------------------------------------------------------------
__global__ __launch_bounds__(128)
void attention_kernel(const float* __restrict__ q, const float* __restrict__ k,
                      const float* __restrict__ v,
                      const float* __restrict__ mclass,
                      const float* __restrict__ fillp,   // nullptr -> 1.0
                      float* __restrict__ out) {
    __shared__ float pbuf[4][16 * 16];
    const int wave = threadIdx.x >> 5;
    const int lane = threadIdx.x & 31;
    const int half = lane >> 4;
    const int l16  = lane & 15;

    const int gw = blockIdx.x * 4 + wave;      // 0 .. 4095
    const int qt = gw & 127;                   // query tile
    const int bh = gw >> 7;                    // 0 .. 31
    const int b  = bh >> 3;
    const int hh = bh & 7;

    const float fill = fillp ? *fillp : 1.0f;
    const size_t bo = (size_t)b * SDIM * EDIM + (size_t)hh * DH;
    const float* qb = q + bo;
    const float* kb = k + bo;
    const float* vb = v + bo;
    const float* mc = mclass + (size_t)b * SDIM;

    // Load 16x64 q tile in A-format (held in regs across the key loop).
    v2f qa[16];
    {
        const float* qr = qb + (size_t)(qt * 16 + l16) * EDIM + half * 2;
#pragma unroll
        for (int d = 0; d < 16; ++d) qa[d] = *(const v2f*)(qr + d * 4);
    }

    float rmax[8], rsum[8];
#pragma unroll
    for (int j = 0; j < 8; ++j) { rmax[j] = -INFINITY; rsum[j] = 0.f; }
    v8f o[4];
#pragma unroll
    for (int t = 0; t < 4; ++t) o[t] = zero8();

    for (int kt = 0; kt < SDIM / 16; ++kt) {
        const int key = kt * 16 + l16;
        const float m = mc[key];
        const float mval = (m == 0.f) ? 0.f : ((m == 2.f) ? fill : 1.f);

        // S = q @ k^T  (16x16)
        v8f sc = zero8();
        const float* kr = kb + (size_t)key * EDIM + half * 2;
#pragma unroll
        for (int d = 0; d < 16; ++d) {
            v2f bkv = *(const v2f*)(kr + d * 4);
            sc = wmma4_f32(qa[d], bkv, sc);
        }

        // mask + online softmax (row reductions over 16 lanes of each half)
#pragma unroll
        for (int j = 0; j < 8; ++j) {
            float x = (mval != 0.f) ? sc[j] * (1.f / (float)DH) * mval
                                    : -INFINITY;
            float mx = x;
#pragma unroll
            for (int off = 1; off < 16; off <<= 1)
                mx = fmaxf(mx, __shfl_xor(mx, off, 32));
            float nmax = fmaxf(rmax[j], mx);
            float corr = (rmax[j] == -INFINITY) ? 0.f : __expf(rmax[j] - nmax);
            float p = (mval != 0.f) ? __expf(x - nmax) : 0.f;
            float ps = p;
#pragma unroll
            for (int off = 1; off < 16; off <<= 1)
                ps += __shfl_xor(ps, off, 32);
            rsum[j] = rsum[j] * corr + ps;
            rmax[j] = nmax;
#pragma unroll
            for (int t = 0; t < 4; ++t) o[t][j] *= corr;
            pbuf[wave][(j + half * 8) * 16 + l16] = p;   // C-layout row
        }
        __syncthreads();   // P(LDS) visible for A-format reads

        // O += P(16x16) @ V(16x64)
#pragma unroll
        for (int kj = 0; kj < 16; kj += 4) {
            v2f a;
            a.x = pbuf[wave][l16 * 16 + kj + half * 2];
            a.y = pbuf[wave][l16 * 16 + kj + half * 2 + 1];
#pragma unroll
            for (int t = 0; t < 4; ++t) {
                const float* vp = vb + (size_t)(kt * 16 + kj + half * 2) * EDIM
                                   + t * 16 + l16;
                v2f bv;
                bv.x = vp[0];
                bv.y = vp[EDIM];
                o[t] = wmma4_f32(a, bv, o[t]);
            }
        }
        __syncthreads();
    }

    // normalize + store
#pragma unroll
    for (int t = 0; t < 4; ++t) {
#pragma unroll
        for (int j = 0; j < 8; ++j) {
            int row = qt * 16 + j + half * 8;
            out[bo + (size_t)row * EDIM + t * 16 + l16] = o[t][j] / rsum[j];
        }
    }
}

// ---------------------------------------------------------------------------
// LayerNorm over E=512. One 256-thread block per row.
// ---------------------------------------------------------------------------
__global__ __launch_bounds__(256)
void layernorm_kernel(const float* __restrict__ in, const float* __restrict__ sc,
                      const float* __restrict__ bi, float* __restrict__ outp) {
    __shared__ float red[8];
    const int row = blockIdx.x;
    const int t = threadIdx.x;
    const int lane = t & 31, wv = t >> 5;
    const float* r = in + (size_t)row * EDIM;
    float x0 = r[t], x1 = r[t + 256];

    float sum = x0 + x1;
#pragma unroll
    for (int off = 16; off > 0; off >>= 1) sum += __shfl_xor(sum, off, 32);
    if (lane == 0) red[wv] = sum;
    __syncthreads();
    float mean = 0.f;
#pragma unroll
    for (int i = 0; i < 8; ++i) mean += red[i];
    mean *= (1.f / (float)EDIM);
    __syncthreads();

    float d0 = x0 - mean, d1 = x1 - mean;
    float vs = d0 * d0 + d1 * d1;
#pragma unroll
    for (int off = 16; off > 0; off >>= 1) vs += __shfl_xor(vs, off, 32);
    if (lane == 0) red[wv] = vs;
    __syncthreads();
    float var = 0.f;
#pragma unroll
    for (int i = 0; i < 8; ++i) var += red[i];
    var *= (1.f / (float)EDIM);
    float inv = rsqrtf(var + 1e-5f);

    outp[(size_t)row * EDIM + t]       = d0 * inv * sc[t] + bi[t];
    outp[(size_t)row * EDIM + t + 256] = d1 * inv * sc[t + 256] + bi[t + 256];
}

// ---------------------------------------------------------------------------
// Final head: gather masked tokens, fc (512 x 35), split L1 / CE.
// ---------------------------------------------------------------------------
#define OUT_L1    0
#define OUT_CE    15360
#define OUT_LABEL 35840
#define OUT_H     52224
#define OUT_MASK  4246528

__global__ void head_kernel(const float* __restrict__ h,
                            const int* __restrict__ masked_pos,
                            const float* __restrict__ fcw,
                            const float* __restrict__ fcb,
                            float* __restrict__ out) {
    int idx = blockIdx.x * blockDim.x + threadIdx.x;
    if (idx >= BDIM * NMASK * 35) return;
    int r = idx / 35, c = idx % 35;
    int b = r >> 8;
    int p = masked_pos[r];
    const float* hr = h + ((size_t)b * SDIM + p) * EDIM;
    float acc = fcb[c];
    for (int e = 0; e < EDIM; ++e) acc += hr[e] * fcw[(size_t)e * 35 + c];
    if (c < 15) out[OUT_L1 + r * 15 + c]        = acc;
    else        out[OUT_CE + r * 20 + (c - 15)] = acc;
}

__global__ void copy_kernel(const float* __restrict__ src,
                            float* __restrict__ dst, int n) {
    int i = blockIdx.x * blockDim.x + threadIdx.x;
    if (i < n) dst[i] = src[i];
}

// ---------------------------------------------------------------------------
// Orchestration
// ---------------------------------------------------------------------------
extern "C" void kernel_launch(void* const* d_in, const int* in_sizes, int n_in,
                              void* d_out, int out_size, void* d_ws,
                              size_t ws_size, hipStream_t stream) {
    (void)in_sizes; (void)n_in; (void)out_size; (void)ws_size;

    const float* x          = (const float*)d_in[0];
    const unsigned char* mk = (const unsigned char*)d_in[1];
    const int* masked_pos   = (const int*)d_in[2];
    const float* arg1       = (const float*)d_in[3];
    const float* arg2       = (const float*)d_in[4];
    const float* emb_table  = (const float*)d_in[5];
    const float* pe         = (const float*)d_in[6];
    const float* Wq = (const float*)d_in[7];
    const float* bq = (const float*)d_in[8];
    const float* Wk = (const float*)d_in[9];
    const float* bk = (const float*)d_in[10];
    const float* Wv = (const float*)d_in[11];
    const float* bv = (const float*)d_in[12];
    const float* Wo = (const float*)d_in[13];
    const float* bo = (const float*)d_in[14];
    const float* ln1s = (const float*)d_in[15];
    const float* ln1b = (const float*)d_in[16];
    const float* W1 = (const float*)d_in[17];
    const float* b1 = (const float*)d_in[18];
    const float* W2 = (const float*)d_in[19];
    const float* b2 = (const float*)d_in[20];
    const float* ln2s = (const float*)d_in[21];
    const float* ln2b = (const float*)d_in[22];
    const float* fcw = (const float*)d_in[23];
    const float* fcb = (const float*)d_in[24];
    float* out = (float*)d_out;

    // workspace layout (floats)
    const size_t TSZ = (size_t)MROWS * EDIM;   // 4,194,304
    float* H   = (float*)d_ws;
    float* Q   = H  + TSZ;
    float* Kb  = Q  + TSZ;
    float* Vb  = Kb + TSZ;
    float* AO  = Vb + TSZ;
    float* T   = AO + TSZ;
    float* FF1 = Q;                // alias: spans Q..AO (4*TSZ == MROWS*DFF)
    float* MC  = T + TSZ;          // mclass, MROWS floats

    // ---- preprocess ----
    mask_init_kernel<<<(BDIM * SDIM + 255) / 256, 256, 0, stream>>>(
        mk, MC, out + OUT_MASK);
    scatter_masked_kernel<<<(BDIM * NMASK + 255) / 256, 256, 0, stream>>>(
        x, masked_pos, MC, out + OUT_LABEL);
    build_h0_kernel<<<MROWS, 128, 0, stream>>>(x, MC, emb_table, pe, H);

    // ---- transformer layers ----
    const dim3 gE(EDIM / 64, MROWS / 128);    // N=512
    const dim3 gF(DFF / 64, MROWS / 128);     // N=2048
    for (int i = 0; i < 4; ++i) {
        const float* Wqi = Wq + (size_t)i * EDIM * EDIM;
        const float* Wki = Wk + (size_t)i * EDIM * EDIM;
        const float* Wvi = Wv + (size_t)i * EDIM * EDIM;
        const float* Woi = Wo + (size_t)i * EDIM * EDIM;
        const float* W1i = W1 + (size_t)i * EDIM * DFF;
        const float* W2i = W2 + (size_t)i * DFF * EDIM;

        gemm_f32_kernel<false, false><<<gE, 256, 0, stream>>>(
            H, Wqi, bq + i * EDIM, nullptr, Q, MROWS, EDIM, EDIM);
        gemm_f32_kernel<false, false><<<gE, 256, 0, stream>>>(
            H, Wki, bk + i * EDIM, nullptr, Kb, MROWS, EDIM, EDIM);
        gemm_f32_kernel<false, false><<<gE, 256, 0, stream>>>(
            H, Wvi, bv + i * EDIM, nullptr, Vb, MROWS, EDIM, EDIM);

        const float* fillp = (i == 0) ? arg1 : (i == 1) ? arg2 : nullptr;
        attention_kernel<<<(BDIM * NHEAD * (SDIM / 16)) / 4, 128, 0, stream>>>(
            Q, Kb, Vb, MC, fillp, AO);

        gemm_f32_kernel<false, true><<<gE, 256, 0, stream>>>(
            AO, Woi, bo + i * EDIM, H, T, MROWS, EDIM, EDIM);
        layernorm_kernel<<<MROWS, 256, 0, stream>>>(
            T, ln1s + i * EDIM, ln1b + i * EDIM, H);

        gemm_f32_kernel<true, false><<<gF, 256, 0, stream>>>(
            H, W1i, b1 + i * DFF, nullptr, FF1, MROWS, DFF, EDIM);
        gemm_f32_kernel<false, true><<<gE, 256, 0, stream>>>(
            FF1, W2i, b2 + i * EDIM, H, T, MROWS, EDIM, DFF);
        layernorm_kernel<<<MROWS, 256, 0, stream>>>(
            T, ln2s + i * EDIM, ln2b + i * EDIM, H);
    }

    // ---- outputs ----
    copy_kernel<<<(int)((TSZ + 255) / 256), 256, 0, stream>>>(
        H, out + OUT_H, (int)TSZ);
    head_kernel<<<(BDIM * NMASK * 35 + 255) / 256, 256, 0, stream>>>(
        H, masked_pos, fcw, fcb, out);
}